// EnhancedPointNet2_50689204027591
// MI455X (gfx1250) — compile-verified
//
#include <hip/hip_runtime.h>

typedef float v2f __attribute__((ext_vector_type(2)));
typedef float v8f __attribute__((ext_vector_type(8)));

#define BN_INV 0.9999950000374996f /* 1/sqrt(1+1e-5) */
#define MAX_K   1536
#define MAX_KS  (MAX_K + 4)        /* LDS column stride: +4 to avoid bank conflicts */

// ---------------------------------------------------------------------------
// Farthest point sampling: one workgroup per batch, dist array in LDS.
// ---------------------------------------------------------------------------
__global__ void __launch_bounds__(256)
pn2_fps_kernel(const float* __restrict__ xyz, int N, int npoint,
               float* __restrict__ new_xyz)
{
    __shared__ float sdist[8192];
    __shared__ float rv[256];
    __shared__ int   ri[256];
    __shared__ int   s_far;
    const int b = blockIdx.x;
    const int tid = threadIdx.x;
    const float* P = xyz + (size_t)b * N * 3;
    float* Q = new_xyz + (size_t)b * npoint * 3;
    for (int i = tid; i < N; i += 256) sdist[i] = 1e10f;
    int far = 0;
    __syncthreads();
    for (int it = 0; it < npoint; ++it) {
        if (tid == 0) {
            Q[it * 3 + 0] = P[far * 3 + 0];
            Q[it * 3 + 1] = P[far * 3 + 1];
            Q[it * 3 + 2] = P[far * 3 + 2];
        }
        if (it + 1 == npoint) break;
        const float cx = P[far * 3 + 0];
        const float cy = P[far * 3 + 1];
        const float cz = P[far * 3 + 2];
        float bv = -1.0f; int bi = 0;
        for (int i = tid; i < N; i += 256) {
            float dx = P[i * 3 + 0] - cx;
            float dy = P[i * 3 + 1] - cy;
            float dz = P[i * 3 + 2] - cz;
            float d = fminf(sdist[i], dx * dx + dy * dy + dz * dz);
            sdist[i] = d;
            if (d > bv) { bv = d; bi = i; }
        }
        rv[tid] = bv; ri[tid] = bi;
        __syncthreads();
        for (int s = 128; s > 0; s >>= 1) {
            if (tid < s) {
                if (rv[tid + s] > rv[tid] ||
                    (rv[tid + s] == rv[tid] && ri[tid + s] < ri[tid])) {
                    rv[tid] = rv[tid + s]; ri[tid] = ri[tid + s];
                }
            }
            __syncthreads();
        }
        if (tid == 0) s_far = ri[0];
        __syncthreads();
        far = s_far;
        __syncthreads();
    }
}

// ---------------------------------------------------------------------------
// Ball query: first <=k indices (ascending) within radius, padded with first.
// ---------------------------------------------------------------------------
__global__ void pn2_ball_query_kernel(const float* __restrict__ xyz,
                                      const float* __restrict__ new_xyz,
                                      int N, int np, int total,
                                      float r2, int k, int* __restrict__ idx)
{
    int t = blockIdx.x * 256 + threadIdx.x;
    if (t >= total) return;
    int b = t / np, q = t % np;
    const float* P = xyz + (size_t)b * N * 3;
    const float* c = new_xyz + ((size_t)b * np + q) * 3;
    float cx = c[0], cy = c[1], cz = c[2];
    int* out = idx + (size_t)t * k;
    int cnt = 0;
    for (int j = 0; j < N && cnt < k; ++j) {
        float dx = P[j * 3 + 0] - cx;
        float dy = P[j * 3 + 1] - cy;
        float dz = P[j * 3 + 2] - cz;
        if (dx * dx + dy * dy + dz * dz <= r2) out[cnt++] = j;
    }
    int first = (cnt > 0) ? out[0] : 0;
    for (int j = cnt; j < k; ++j) out[j] = first;
}

// ---------------------------------------------------------------------------
// Grouping: build rows of [xyz_local(3) | feats(C)] for the grouped points.
// ---------------------------------------------------------------------------
__global__ void pn2_group_kernel(const float* __restrict__ xyz,
                                 const float* __restrict__ feats, int C,
                                 const float* __restrict__ new_xyz,
                                 const int* __restrict__ idx,
                                 int N, int np, int k, int total,
                                 float* __restrict__ gin)
{
    int t = blockIdx.x * 256 + threadIdx.x;
    if (t >= total) return;
    int Cin = 3 + C;
    int c = t % Cin;
    int row = t / Cin;              // row = (b*np + q)*k + j
    int bq = row / k;
    int q = bq % np;
    int b = bq / np;
    int ptidx = idx[row];
    float v;
    if (c < 3)
        v = xyz[((size_t)b * N + ptidx) * 3 + c] -
            new_xyz[((size_t)b * np + q) * 3 + c];
    else
        v = feats[((size_t)b * N + ptidx) * C + (c - 3)];
    gin[(size_t)row * Cin + c] = v;
}

// ---------------------------------------------------------------------------
// WMMA fp32 GEMM: out[rows x N] = A[rows x K] * W[K x N] (+bias)(*BN)(ReLU)
//
// Block = 8 waves; each wave owns one 16-row M-tile and TWO 16-col N-tiles
// (reusing each A fragment for two WMMAs). The K x 32 weight slice is staged
// in LDS (column-major, zero-padded to Kpad = 16n). Bulk K loop is fully
// branch- and mask-free; only the final <=3-element K tail is masked.
// ---------------------------------------------------------------------------
__global__ void __launch_bounds__(256)
pn2_gemm_bn_relu_kernel(const float* __restrict__ A, int rows, int K,
                        const float* __restrict__ W, int N,
                        const float* __restrict__ bias,
                        const float* __restrict__ g,
                        const float* __restrict__ beta,
                        int fuse, float* __restrict__ out)
{
    __shared__ float Wl[32 * MAX_KS];   // ~197 KB, within 320 KB/WGP

    const int tid  = threadIdx.x;
    const int wid  = tid >> 5;
    const int lane = tid & 31;
    const int tilesM = (rows + 15) >> 4;
    const int nb32 = (N + 31) >> 5;
    const int tn   = blockIdx.x % nb32;          // 32-column super-tile
    const int mblk = blockIdx.x / nb32;
    const int tm   = mblk * 8 + wid;

    const int Kpad = (K + 15) & ~15;
    const int Ks   = Kpad + 4;
    const int cbase = tn * 32;

    // ---- stage zero-padded 32-col weight slice, column-major Wl[col*Ks+k] ----
    for (int idx = tid; idx < 32 * Kpad; idx += 256) {
        int col = idx & 31;
        int k = idx >> 5;
        float v = 0.0f;
        if (k < K && cbase + col < N) v = W[(size_t)k * N + cbase + col];
        Wl[col * Ks + k] = v;
    }
    __syncthreads();

    if (tm >= tilesM) return;

    const int half = lane >> 4;        // A/B lane halves: K {kb,kb+1} vs {kb+2,kb+3}
    const int l = lane & 15;
    const int arow = tm * 16 + l;
    const int amin = arow < rows ? arow : rows - 1;   // clamp: garbage rows unstored
    const float* __restrict__ Ap  = A + (size_t)amin * K;
    const float* __restrict__ Bl0 = &Wl[l * Ks];
    const float* __restrict__ Bl1 = &Wl[(l + 16) * Ks];

    v8f acc0 = {};
    v8f acc1 = {};

    auto step = [&](int kb) {
        const int ka = kb + 2 * half;
        v2f a;  a.x  = Ap[ka];       a.y  = Ap[ka + 1];
        v2f b0; b0.x = Bl0[ka];      b0.y = Bl0[ka + 1];
        v2f b1; b1.x = Bl1[ka];      b1.y = Bl1[ka + 1];
        acc0 = __builtin_amdgcn_wmma_f32_16x16x4_f32(
            false, a, false, b0, (short)0, acc0, false, false);
        acc1 = __builtin_amdgcn_wmma_f32_16x16x4_f32(
            false, a, false, b1, (short)0, acc1, false, false);
    };

    int kb = 0;
    const int K16 = K & ~15;
    for (; kb < K16; kb += 16) {                 // bulk: mask-free, 8 wmma/iter
        __builtin_prefetch(Ap + kb + 96, 0, 1);
        step(kb); step(kb + 4); step(kb + 8); step(kb + 12);
    }
    const int K4 = K & ~3;
    for (; kb < K4; kb += 4) step(kb);           // 4-wide remainder, mask-free
    if (kb < K) {                                // <=3-element tail, masked A
        const int ka = kb + 2 * half;
        float a0 = Ap[ka];
        float a1 = Ap[ka + 1];
        if (ka >= K)     a0 = 0.0f;
        if (ka + 1 >= K) a1 = 0.0f;
        v2f a;  a.x  = a0;        a.y  = a1;
        v2f b0; b0.x = Bl0[ka];   b0.y = Bl0[ka + 1];   // pad region is zero
        v2f b1; b1.x = Bl1[ka];   b1.y = Bl1[ka + 1];
        acc0 = __builtin_amdgcn_wmma_f32_16x16x4_f32(
            false, a, false, b0, (short)0, acc0, false, false);
        acc1 = __builtin_amdgcn_wmma_f32_16x16x4_f32(
            false, a, false, b1, (short)0, acc1, false, false);
    }

    auto store_tile = [&](const v8f& acc, int col) {
        if (col >= N) return;
        float sb = bias ? bias[col] : 0.0f;
        float sg = 1.0f, sbe = 0.0f;
        if (fuse) { sg = g[col] * BN_INV; sbe = beta[col]; }
        for (int i = 0; i < 8; ++i) {
            int m = tm * 16 + i + 8 * half;      // C/D layout: VGPR i, upper half +8
            if (m < rows) {
                float v = acc[i] + sb;
                if (fuse) v = fmaxf(v * sg + sbe, 0.0f);
                out[(size_t)m * N + col] = v;
            }
        }
    };
    store_tile(acc0, cbase + l);
    store_tile(acc1, cbase + 16 + l);
}

// ---------------------------------------------------------------------------
// Max over the nsample axis; writes into concat destination at channel offset.
// ---------------------------------------------------------------------------
__global__ void pn2_maxpool_kernel(const float* __restrict__ src,
                                   int k, int C, int total,
                                   float* __restrict__ dst, int Cdst, int coff)
{
    int t = blockIdx.x * 256 + threadIdx.x;
    if (t >= total) return;
    int c = t % C;
    int bq = t / C;
    const float* s = src + (size_t)bq * k * C + c;
    float m = s[0];
    for (int j = 1; j < k; ++j) m = fmaxf(m, s[(size_t)j * C]);
    dst[(size_t)bq * Cdst + coff + c] = m;
}

// ---------------------------------------------------------------------------
// 3-NN with inverse-distance weights (normalized). One thread per point.
// ---------------------------------------------------------------------------
__global__ void pn2_knn3_kernel(const float* __restrict__ xyz1, int n1,
                                const float* __restrict__ xyz2, int n2,
                                int total, int* __restrict__ idx3,
                                float* __restrict__ w3)
{
    int t = blockIdx.x * 256 + threadIdx.x;
    if (t >= total) return;
    int b = t / n1;
    const float* p = xyz1 + (size_t)t * 3;
    float px = p[0], py = p[1], pz = p[2];
    const float* P2 = xyz2 + (size_t)b * n2 * 3;
    float d0 = 1e30f, d1 = 1e30f, d2 = 1e30f;
    int i0 = 0, i1 = 0, i2 = 0;
    for (int j = 0; j < n2; ++j) {
        float dx = P2[j * 3 + 0] - px;
        float dy = P2[j * 3 + 1] - py;
        float dz = P2[j * 3 + 2] - pz;
        float d = dx * dx + dy * dy + dz * dz;
        if (d < d0)      { d2 = d1; i2 = i1; d1 = d0; i1 = i0; d0 = d; i0 = j; }
        else if (d < d1) { d2 = d1; i2 = i1; d1 = d; i1 = j; }
        else if (d < d2) { d2 = d; i2 = j; }
    }
    float w0 = 1.0f / (d0 + 1e-8f);
    float w1 = 1.0f / (d1 + 1e-8f);
    float w2 = 1.0f / (d2 + 1e-8f);
    float s = w0 + w1 + w2;
    idx3[t * 3 + 0] = i0; idx3[t * 3 + 1] = i1; idx3[t * 3 + 2] = i2;
    w3[t * 3 + 0] = w0 / s; w3[t * 3 + 1] = w1 / s; w3[t * 3 + 2] = w2 / s;
}

// ---------------------------------------------------------------------------
// Interpolate features from coarser level + concat with skip features.
// ---------------------------------------------------------------------------
__global__ void pn2_interp_concat_kernel(const float* __restrict__ p1, int C1,
                                         const float* __restrict__ p2, int C2,
                                         const int* __restrict__ idx3,
                                         const float* __restrict__ w3,
                                         int n1, int n2, int total,
                                         float* __restrict__ gin)
{
    int t = blockIdx.x * 256 + threadIdx.x;
    if (t >= total) return;
    int Cin = C1 + C2;
    int c = t % Cin;
    int r = t / Cin;        // r = b*n1 + i
    int b = r / n1;
    float v;
    if (c < C1) {
        v = p1[(size_t)r * C1 + c];
    } else {
        int cc = c - C1;
        const float* F = p2 + (size_t)b * n2 * C2;
        v = w3[r * 3 + 0] * F[(size_t)idx3[r * 3 + 0] * C2 + cc]
          + w3[r * 3 + 1] * F[(size_t)idx3[r * 3 + 1] * C2 + cc]
          + w3[r * 3 + 2] * F[(size_t)idx3[r * 3 + 2] * C2 + cc];
    }
    gin[(size_t)r * Cin + c] = v;
}

// ---------------------------------------------------------------------------
// (B, N, C) rows -> (B, C, N) output
// ---------------------------------------------------------------------------
__global__ void pn2_transpose_out_kernel(const float* __restrict__ h,
                                         int N, int C, int total,
                                         float* __restrict__ out)
{
    int t = blockIdx.x * 256 + threadIdx.x;
    if (t >= total) return;
    int n = t % N;
    int bc = t / N;
    int c = bc % C;
    int b = bc / C;
    out[t] = h[((size_t)b * N + n) * C + c];
}

// ===========================================================================
// Host side
// ===========================================================================
namespace {

struct Layer { const float *W, *b, *g, *beta; };

inline int cdiv(int a, int b) { return (a + b - 1) / b; }

void launch_gemm(hipStream_t s, const float* A, int rows, int K,
                 const Layer& L, int N, int fuse, float* out)
{
    int tilesM = cdiv(rows, 16);
    int blocks = cdiv(tilesM, 8) * cdiv(N, 32);
    pn2_gemm_bn_relu_kernel<<<blocks, 256, 0, s>>>(A, rows, K, L.W, N,
                                                   L.b, L.g, L.beta, fuse, out);
}

// Runs the MLP chain; returns pointer to final output buffer.
const float* mlp_chain(hipStream_t s, const float* in, int rows, int Cin,
                       const Layer* Ls, const int* chs, int nl,
                       float* ping, float* pong, float* final_dst /*or null*/)
{
    const float* cur = in;
    float* dst = ping;
    int K = Cin;
    for (int i = 0; i < nl; ++i) {
        float* d = (final_dst && i == nl - 1) ? final_dst : dst;
        launch_gemm(s, cur, rows, K, Ls[i], chs[i], 1, d);
        cur = d;
        K = chs[i];
        dst = (cur == ping) ? pong : ping;
    }
    return cur;
}

void run_sa(hipStream_t s, const float* in_xyz, int Nin,
            const float* in_feats, int C, int np,
            const float* radii, const int* ks,
            const Layer sc[2][3], const int* chs,
            int* ballIdx, float* gin, float* ping, float* pong,
            float* out_xyz, float* out_feat)
{
    const int B = 4;
    pn2_fps_kernel<<<B, 256, 0, s>>>(in_xyz, Nin, np, out_xyz);
    int lastC = chs[2];
    for (int sc_i = 0; sc_i < 2; ++sc_i) {
        float r = radii[sc_i];
        int k = ks[sc_i];
        int nq = B * np;
        pn2_ball_query_kernel<<<cdiv(nq, 256), 256, 0, s>>>(
            in_xyz, out_xyz, Nin, np, nq, r * r, k, ballIdx);
        int rows = B * np * k;
        int Cin = 3 + C;
        int gtotal = rows * Cin;
        pn2_group_kernel<<<cdiv(gtotal, 256), 256, 0, s>>>(
            in_xyz, in_feats, C, out_xyz, ballIdx, Nin, np, k, gtotal, gin);
        const float* last =
            mlp_chain(s, gin, rows, Cin, sc[sc_i], chs, 3, ping, pong, nullptr);
        int ptotal = B * np * lastC;
        pn2_maxpool_kernel<<<cdiv(ptotal, 256), 256, 0, s>>>(
            last, k, lastC, ptotal, out_feat, 2 * lastC, sc_i * lastC);
    }
}

void run_fp(hipStream_t s, const float* xyz1, int n1, const float* xyz2, int n2,
            const float* p1, int C1, const float* p2, int C2,
            const Layer* Ls, const int* chs, int nl,
            int* knnIdx, float* knnW, float* gin, float* ping, float* pong,
            float* out)
{
    const int B = 4;
    int npts = B * n1;
    pn2_knn3_kernel<<<cdiv(npts, 256), 256, 0, s>>>(xyz1, n1, xyz2, n2, npts,
                                                    knnIdx, knnW);
    int Cin = C1 + C2;
    int itotal = npts * Cin;
    pn2_interp_concat_kernel<<<cdiv(itotal, 256), 256, 0, s>>>(
        p1, C1, p2, C2, knnIdx, knnW, n1, n2, itotal, gin);
    mlp_chain(s, gin, npts, Cin, Ls, chs, nl, ping, pong, out);
}

} // namespace

extern "C" void kernel_launch(void* const* d_in, const int* in_sizes, int n_in,
                              void* d_out, int out_size, void* d_ws, size_t ws_size,
                              hipStream_t stream)
{
    (void)in_sizes; (void)n_in; (void)ws_size;
    const int B = 4, N = 8192;

    // ---- unpack inputs (setup_inputs insertion order) ----
    int p = 0;
    const float* xyz    = (const float*)d_in[p++];
    const float* colors = (const float*)d_in[p++];
    auto take = [&](Layer& L) {
        L.W    = (const float*)d_in[p++];
        L.b    = (const float*)d_in[p++];
        L.g    = (const float*)d_in[p++];
        L.beta = (const float*)d_in[p++];
    };
    Layer sa1L[2][3], sa2L[2][3], sa3L[2][3];
    for (int s = 0; s < 2; ++s) for (int l = 0; l < 3; ++l) take(sa1L[s][l]);
    for (int s = 0; s < 2; ++s) for (int l = 0; l < 3; ++l) take(sa2L[s][l]);
    for (int s = 0; s < 2; ++s) for (int l = 0; l < 3; ++l) take(sa3L[s][l]);
    Layer fp3L[2], fp2L[2], fp1L[3], h1;
    for (int l = 0; l < 2; ++l) take(fp3L[l]);
    for (int l = 0; l < 2; ++l) take(fp2L[l]);
    for (int l = 0; l < 3; ++l) take(fp1L[l]);
    take(h1);
    Layer h2;
    h2.W = (const float*)d_in[p++];
    h2.b = (const float*)d_in[p++];
    h2.g = nullptr; h2.beta = nullptr;

    // ---- workspace carve-up ----
    char* base = (char*)d_ws;
    size_t off = 0;
    auto alloc = [&](size_t bytes) -> void* {
        off = (off + 255) & ~(size_t)255;
        void* r = base + off;
        off += bytes;
        return r;
    };
    float* l1_xyz  = (float*)alloc((size_t)B * 1024 * 3 * 4);
    float* l2_xyz  = (float*)alloc((size_t)B * 256 * 3 * 4);
    float* l3_xyz  = (float*)alloc((size_t)B * 64 * 3 * 4);
    float* l1_feat = (float*)alloc((size_t)B * 1024 * 256 * 4);
    float* l2_feat = (float*)alloc((size_t)B * 256 * 512 * 4);
    float* l3_feat = (float*)alloc((size_t)B * 64 * 1024 * 4);
    float* fp3_out = (float*)alloc((size_t)B * 256 * 256 * 4);
    float* fp2_out = (float*)alloc((size_t)B * 1024 * 128 * 4);
    float* fp1_out = (float*)alloc((size_t)B * 8192 * 128 * 4);
    int*   ballIdx = (int*)alloc((size_t)B * 1024 * 32 * 4);
    int*   knnIdx  = (int*)alloc((size_t)B * 8192 * 3 * 4);
    float* knnW    = (float*)alloc((size_t)B * 8192 * 3 * 4);
    float* gin     = (float*)alloc((size_t)8486912 * 4);      // max grouped input
    float* ping    = (float*)alloc((size_t)16777216 * 4);     // 131072 rows x 128
    float* pong    = (float*)alloc((size_t)16777216 * 4);

    // ---- SA levels (multi-scale grouping) ----
    const float r1[2] = {0.1f, 0.2f};
    const float r2_[2] = {0.2f, 0.4f};
    const float r3[2] = {0.4f, 0.8f};
    const int ks[2] = {16, 32};
    const int ch1[3] = {64, 64, 128};
    const int ch2[3] = {128, 128, 256};
    const int ch3[3] = {256, 256, 512};

    run_sa(stream, xyz, N, colors, 6, 1024, r1, ks, sa1L, ch1,
           ballIdx, gin, ping, pong, l1_xyz, l1_feat);
    run_sa(stream, l1_xyz, 1024, l1_feat, 256, 256, r2_, ks, sa2L, ch2,
           ballIdx, gin, ping, pong, l2_xyz, l2_feat);
    run_sa(stream, l2_xyz, 256, l2_feat, 512, 64, r3, ks, sa3L, ch3,
           ballIdx, gin, ping, pong, l3_xyz, l3_feat);

    // ---- FP levels ----
    const int chfp3[2] = {512, 256};
    const int chfp2[2] = {256, 128};
    const int chfp1[3] = {128, 128, 128};
    run_fp(stream, l2_xyz, 256, l3_xyz, 64, l2_feat, 512, l3_feat, 1024,
           fp3L, chfp3, 2, knnIdx, knnW, gin, ping, pong, fp3_out);
    run_fp(stream, l1_xyz, 1024, l2_xyz, 256, l1_feat, 256, fp3_out, 256,
           fp2L, chfp2, 2, knnIdx, knnW, gin, ping, pong, fp2_out);
    run_fp(stream, xyz, N, l1_xyz, 1024, nullptr, 0, fp2_out, 128,
           fp1L, chfp1, 3, knnIdx, knnW, gin, ping, pong, fp1_out);

    // ---- heads ----
    int rows = B * N;
    launch_gemm(stream, fp1_out, rows, 128, h1, 128, 1, ping);
    launch_gemm(stream, ping, rows, 128, h2, 8, 0, pong);

    int ttotal = B * 8 * N;  // == out_size
    (void)out_size;
    pn2_transpose_out_kernel<<<cdiv(ttotal, 256), 256, 0, stream>>>(
        pong, N, 8, ttotal, (float*)d_out);
}